// VAE_32650341384226
// MI455X (gfx1250) — compile-verified
//
#include <hip/hip_runtime.h>

// ---------------- problem constants ----------------
#define F_      32
#define C_      64
#define DSF     4
#define ORD     8
#define T_RAW   16384
#define PAD     64
#define TPADDED (T_RAW + PAD)   // 16448
#define TP      (TPADDED / DSF) // 4112  (= 257*16)
#define NE      (TP - ORD)      // 4104
#define ND      (NE - ORD)      // 4096

typedef float v2f __attribute__((ext_vector_type(2)));
typedef float v4f __attribute__((ext_vector_type(4)));
typedef float v8f __attribute__((ext_vector_type(8)));

// =====================================================================
// Generic WMMA kernel: for each slice g, D[row,n] (+)= sum_K W[row,K]*B[g,K,n+shift]
//   B element (g,K,n): Bin[g*sgIn + K*skIn + n + shift]
//   D element (g,row,n): Dout[g*sgOut + row*smOut + n]         (dup4==0)
//   or out[(rowOffset + g*sgOut + row*smOut)*16384 + 4n+{0..3}] (dup4==1, splat NT)
// W is Mdim x 32 row-major (Mdim = 16*Mtiles). One wave per (g,mt,nt) tile.
// Edge handling: out-of-range columns are CLAMPED on load (columns of a WMMA
// are independent, so garbage only reaches columns we never store) and the
// store is guarded. This keeps EXEC all-ones around the WMMA chain with no
// divergent load branches.
// =====================================================================
__global__ void wmma_slicemm(const float* __restrict__ W,
                             const float* __restrict__ Bin,
                             float* __restrict__ Dout,
                             int G, int Mtiles, int Ntiles,
                             int sgIn, int skIn, int shift,
                             int sgOut, int smOut,
                             int Nvalid, int accumulate,
                             const float* __restrict__ alphaPtr,
                             int dup4, int rowOffset)
{
    const int lane  = threadIdx.x & 31;
    const int wave  = blockIdx.x * (blockDim.x >> 5) + (threadIdx.x >> 5);
    const int tiles = G * Mtiles * Ntiles;
    if (wave >= tiles) return;  // wave-uniform exit, EXEC stays all-ones for WMMA

    const int nt = wave % Ntiles;
    const int mt = (wave / Ntiles) % Mtiles;
    const int g  = wave / (Ntiles * Mtiles);

    const int lhalf = lane >> 4;       // 0 | 1
    const int l15   = lane & 15;
    const int n     = nt * 16 + l15;                  // N (time) coordinate
    const bool nOK  = (n < Nvalid);
    const int nIn   = nOK ? n : (Nvalid - 1);         // clamped (in-bounds) column

    // ---- C accumulator (16x16 f32 C/D layout: reg r -> M = mt*16 + r + 8*lhalf)
    v8f acc;
    if (accumulate && !dup4) {
        const float* __restrict__ cp = Dout + (size_t)g * sgOut + (size_t)(mt * 16 + 8 * lhalf) * smOut + nIn;
#pragma unroll
        for (int r = 0; r < 8; ++r) {
            acc[r] = cp[0];
            cp += smOut;
        }
    } else {
#pragma unroll
        for (int r = 0; r < 8; ++r) acc[r] = 0.0f;
    }

    // A fragment: M = lane&15, K = 4*kk + reg + 2*lhalf  -> consecutive pair (b64)
    const int arow = mt * 16 + l15;
    const float* __restrict__ wp = W + (size_t)arow * 32 + 2 * lhalf;

    // B fragment: K = 4*kk + reg + 2*lhalf, column nIn+shift; running pointers
    const size_t skStep = (size_t)skIn;
    const float* __restrict__ bp0 = Bin + (size_t)g * sgIn + (size_t)(nIn + shift)
                                        + (size_t)(2 * lhalf) * skStep;
    const float* __restrict__ bp1 = bp0 + skStep;

    // ---- K = 32 contraction as 8 chained V_WMMA_F32_16X16X4_F32
#pragma unroll
    for (int kk = 0; kk < 8; ++kk) {
        v2f a, b;
        a.x = wp[0];
        a.y = wp[1];
        b.x = bp0[0];
        b.y = bp1[0];
        acc = __builtin_amdgcn_wmma_f32_16x16x4_f32(
                  /*neg_a=*/false, a, /*neg_b=*/false, b,
                  /*c_mod=*/(short)0, acc, /*reuse_a=*/false, /*reuse_b=*/false);
        wp  += 4;
        bp0 += 4 * skStep;
        bp1 += 4 * skStep;
    }

    const float alpha = alphaPtr ? alphaPtr[0] : 1.0f;
#pragma unroll
    for (int r = 0; r < 8; ++r) acc[r] *= alpha;

    if (dup4) {
        // write each column 4x (time upsample), 16B splat NT stores (write-once output)
        if (nOK) {
            const size_t rowBase = (size_t)rowOffset + (size_t)g * sgOut + (size_t)(mt * 16 + 8 * lhalf) * smOut;
            float* dp = Dout + rowBase * (size_t)T_RAW + 4u * (size_t)n;
#pragma unroll
            for (int r = 0; r < 8; ++r) {
                const float v = acc[r];
                v4f v4 = { v, v, v, v };
                __builtin_nontemporal_store(v4, reinterpret_cast<v4f*>(dp));
                dp += (size_t)smOut * T_RAW;
            }
        }
    } else {
        if (nOK) {
            float* dp = Dout + (size_t)g * sgOut + (size_t)(mt * 16 + 8 * lhalf) * smOut + n;
#pragma unroll
            for (int r = 0; r < 8; ++r) {
                dp[0] = acc[r];
                dp += smOut;
            }
        }
    }
}

// =====================================================================
// filt[k][c][C] = k<8 ? ma_w[c][C][k] : (c==C)
// =====================================================================
__global__ void k_build_filt(const float* __restrict__ maw, float* __restrict__ filt)
{
    const int idx = blockIdx.x * blockDim.x + threadIdx.x;
    if (idx >= 9 * F_ * F_) return;
    const int k  = idx >> 10;          // /1024
    const int cc = idx & 1023;
    const int c  = cc >> 5;
    const int Cc = cc & 31;
    filt[idx] = (k < 8) ? maw[(size_t)cc * ORD + k] : (c == Cc ? 1.0f : 0.0f);
}

// =====================================================================
// s = W_enc * y_pad (reflect pad 64 on the left); store |s| and sign(s)
// =====================================================================
__global__ void k_spad(const float* __restrict__ y, const float* __restrict__ W_enc,
                       float* __restrict__ s_abs, float* __restrict__ s_sign)
{
    const int t = blockIdx.x * blockDim.x + threadIdx.x;
    const int f = blockIdx.y;
    if (t >= TPADDED) return;
    const int tt = (t < PAD) ? (PAD - t) : (t - PAD);   // reflect index into y
    float s = 0.0f;
#pragma unroll 8
    for (int c = 0; c < C_; ++c)
        s = fmaf(W_enc[f * C_ + c], y[(size_t)c * T_RAW + tt], s);
    s_abs[(size_t)f * TPADDED + t] = fabsf(s);
    if (t >= PAD) {
        const float sg = (s > 0.0f) ? 1.0f : ((s < 0.0f) ? -1.0f : 0.0f);
        s_sign[(size_t)f * T_RAW + (t - PAD)] = sg;
    }
}

// a_mean[c][m] = mean_d s_abs[c][4m+d]
__global__ void k_amean(const float* __restrict__ s_abs, float* __restrict__ a_mean)
{
    const int m = blockIdx.x * blockDim.x + threadIdx.x;
    const int c = blockIdx.y;
    if (m >= TP) return;
    const float* p = s_abs + (size_t)c * TPADDED + 4u * (size_t)m;
    a_mean[(size_t)c * TP + m] = 0.25f * (p[0] + p[1] + p[2] + p[3]);
}

// a_cov[(c*32+s)][m] = sum_d cent_c*cent_s / 3
__global__ void k_acov(const float* __restrict__ s_abs, const float* __restrict__ a_mean,
                       float* __restrict__ a_cov)
{
    const int m = blockIdx.x * blockDim.x + threadIdx.x;
    const int p = blockIdx.y;               // c*32 + s
    if (m >= TP) return;
    const int c = p >> 5, s = p & 31;
    const float mc = a_mean[(size_t)c * TP + m];
    const float ms = a_mean[(size_t)s * TP + m];
    const float* pc = s_abs + (size_t)c * TPADDED + 4u * (size_t)m;
    const float* ps = s_abs + (size_t)s * TPADDED + 4u * (size_t)m;
    float acc = 0.0f;
#pragma unroll
    for (int d = 0; d < DSF; ++d)
        acc = fmaf(pc[d] - mc, ps[d] - ms, acc);
    a_cov[(size_t)p * TP + m] = acc * (1.0f / 3.0f);
}

// out[o][m] = b[o] + in[o][m+8] + sum_{i,k<8} maw[o][i][k] * in[i][m+k]
__global__ void k_mamean(const float* __restrict__ in, const float* __restrict__ maw,
                         const float* __restrict__ bias, float* __restrict__ outp,
                         int NI, int NO)
{
    const int m = blockIdx.x * blockDim.x + threadIdx.x;
    const int o = blockIdx.y;
    if (m >= NO) return;
    float acc = bias[o] + in[(size_t)o * NI + m + ORD];
    for (int i = 0; i < F_; ++i) {
        const float* ip = in + (size_t)i * NI + m;
        const float* wp = maw + ((size_t)o * F_ + i) * ORD;
#pragma unroll
        for (int k = 0; k < ORD; ++k)
            acc = fmaf(wp[k], ip[k], acc);
    }
    outp[(size_t)o * NO + m] = acc;
}

// out rows 0..63: y_mean[o][t] = sum_i W_dec[o][i] * a_mean_t[i][t/4] * sign[i][t]
__global__ void k_outmean(const float* __restrict__ a_mean_t, const float* __restrict__ sgn,
                          const float* __restrict__ W_dec, float* __restrict__ outp)
{
    const int t = blockIdx.x * blockDim.x + threadIdx.x;
    const int o = blockIdx.y;
    if (t >= T_RAW) return;
    const int m = t >> 2;
    float acc = 0.0f;
#pragma unroll 8
    for (int i = 0; i < F_; ++i)
        acc = fmaf(W_dec[o * F_ + i],
                   a_mean_t[(size_t)i * ND + m] * sgn[(size_t)i * T_RAW + t], acc);
    __builtin_nontemporal_store(acc, outp + (size_t)o * T_RAW + t);
}

// ---------------- host side ----------------
static inline void launch_mm(hipStream_t s, const float* W, const float* B, float* D,
                             int G, int Mt, int Nt,
                             int sgIn, int skIn, int shift,
                             int sgOut, int smOut, int Nvalid,
                             int accum, const float* alphaPtr, int dup4, int rowOffset)
{
    const int tiles  = G * Mt * Nt;
    const int blocks = (tiles + 3) / 4;     // 4 waves / 128-thread block
    hipLaunchKernelGGL(wmma_slicemm, dim3(blocks), dim3(128), 0, s,
                       W, B, D, G, Mt, Nt, sgIn, skIn, shift,
                       sgOut, smOut, Nvalid, accum, alphaPtr, dup4, rowOffset);
}

extern "C" void kernel_launch(void* const* d_in, const int* in_sizes, int n_in,
                              void* d_out, int out_size, void* d_ws, size_t ws_size,
                              hipStream_t stream)
{
    (void)in_sizes; (void)n_in; (void)out_size; (void)ws_size;
    const float* y     = (const float*)d_in[0];
    const float* W_enc = (const float*)d_in[1];
    const float* W_dec = (const float*)d_in[2];
    const float* mawE  = (const float*)d_in[3];
    const float* mabE  = (const float*)d_in[4];
    const float* mawD  = (const float*)d_in[5];
    const float* mabD  = (const float*)d_in[6];
    const float* covs  = (const float*)d_in[7];
    float* out = (float*)d_out;
    float* ws  = (float*)d_ws;

    // workspace layout (floats); a_cov reused as a_cov_t, out1+x_cov reused as P
    float* s_abs    = ws;
    float* s_sign   = s_abs  + (size_t)F_ * TPADDED;
    float* a_mean   = s_sign + (size_t)F_ * T_RAW;
    float* a_cov    = a_mean + (size_t)F_ * TP;
    float* filt_e   = a_cov  + (size_t)F_ * F_ * TP;
    float* filt_d   = filt_e + 9 * F_ * F_;
    float* x_mean   = filt_d + 9 * F_ * F_;
    float* out1     = x_mean + (size_t)F_ * NE;
    float* x_cov    = out1   + (size_t)F_ * F_ * TP;
    float* a_mean_t = x_cov  + (size_t)F_ * F_ * NE;
    float* a_cov_t  = a_cov;   // alias (a_cov dead after encoder cov chain)
    float* Pbuf     = out1;    // alias (needs out1+x_cov span; both dead by then)

    // filters (MA kernels + identity tap), per-k contiguous 32x32
    hipLaunchKernelGGL(k_build_filt, dim3(36), dim3(256), 0, stream, mawE, filt_e);
    hipLaunchKernelGGL(k_build_filt, dim3(36), dim3(256), 0, stream, mawD, filt_d);

    // encoder front: spatial filter + rectifier + per-4 stats
    hipLaunchKernelGGL(k_spad,  dim3((TPADDED + 255) / 256, F_), dim3(256), 0, stream,
                       y, W_enc, s_abs, s_sign);
    hipLaunchKernelGGL(k_amean, dim3((TP + 255) / 256, F_), dim3(256), 0, stream,
                       s_abs, a_mean);
    hipLaunchKernelGGL(k_acov,  dim3((TP + 255) / 256, F_ * F_), dim3(256), 0, stream,
                       s_abs, a_mean, a_cov);

    // encoder mean MA
    hipLaunchKernelGGL(k_mamean, dim3((NE + 255) / 256, F_), dim3(256), 0, stream,
                       a_mean, mawE, mabE, x_mean, TP, NE);

    // encoder cov MA: x_cov[c,s,m] = scaler * sum_k (F_k A_{m+k} F_k^T)[c,s]
    for (int k = 0; k < 9; ++k) {
        // stage1: out1[c,S,t] = sum_C Fe_k[c,C] * a_cov[C,S,t]      (slice g = S)
        launch_mm(stream, filt_e + k * 1024, a_cov, out1,
                  /*G*/F_, /*Mt*/2, /*Nt*/TP / 16,
                  /*sgIn*/TP, /*skIn*/F_ * TP, /*shift*/0,
                  /*sgOut*/TP, /*smOut*/F_ * TP, /*Nvalid*/TP,
                  0, nullptr, 0, 0);
        // stage2: x_cov[c,s,m] += sum_S Fe_k[s,S] * out1[c,S,m+k]   (slice g = c)
        launch_mm(stream, filt_e + k * 1024, out1, x_cov,
                  F_, 2, (NE + 15) / 16,
                  F_ * TP, TP, k,
                  F_ * NE, NE, NE,
                  (k > 0), (k == 8) ? covs : nullptr, 0, 0);
    }

    // decoder mean MA
    hipLaunchKernelGGL(k_mamean, dim3((ND + 255) / 256, F_), dim3(256), 0, stream,
                       x_mean, mawD, mabD, a_mean_t, NE, ND);

    // decoder cov MA: a_cov_t[c,s,m] = sum_k (Fd_k x_cov_{m+k} Fd_k^T)[c,s]
    for (int k = 0; k < 9; ++k) {
        launch_mm(stream, filt_d + k * 1024, x_cov, out1,
                  F_, 2, (NE + 15) / 16,
                  NE, F_ * NE, 0,
                  NE, F_ * NE, NE,
                  0, nullptr, 0, 0);
        launch_mm(stream, filt_d + k * 1024, out1, a_cov_t,
                  F_, 2, ND / 16,
                  F_ * NE, NE, k,
                  F_ * ND, ND, ND,
                  (k > 0), nullptr, 0, 0);
    }

    // output cov transform: Y = W_dec * S * W_dec^T, upsampled x4 into out rows 64..4159
    // stage A: P[o,I,m] = sum_i W_dec[o,i] * a_cov_t[i,I,m]         (slice g = I)
    launch_mm(stream, W_dec, a_cov_t, Pbuf,
              F_, 4, ND / 16,
              ND, F_ * ND, 0,
              ND, F_ * ND, ND,
              0, nullptr, 0, 0);
    // stage B: out[64 + o*64 + O][4m+d] = sum_I W_dec[O,I] * P[o,I,m]  (slice g = o)
    launch_mm(stream, W_dec, Pbuf, out,
              C_, 4, ND / 16,
              F_ * ND, ND, 0,
              /*sgOut(row idx)*/C_, /*smOut*/1, ND,
              0, nullptr, /*dup4*/1, /*rowOffset*/C_);

    // output mean rows 0..63
    hipLaunchKernelGGL(k_outmean, dim3((T_RAW + 255) / 256, C_), dim3(256), 0, stream,
                       a_mean_t, s_sign, W_dec, out);
}